// EmbeddingCollection_84894323573300
// MI455X (gfx1250) — compile-verified
//
#include <hip/hip_runtime.h>

// EmbeddingCollection gather for MI455X (gfx1250, wave32).
// tables: [T=8, V=200000, D=128] fp32  (d_in[0])
// indices: [T=8, NNZ=81920] int        (d_in[1])
// out:    [T, NNZ, D] fp32             (d_out)
//
// Pure bandwidth-bound gather: ~673 MB total traffic -> ~29 us floor at
// 23.3 TB/s. Zero FLOPs -> WMMA inapplicable. Optimal CDNA5 path:
//  - one wave32 moves one 128-float row as 32 lanes x float4 (512B coalesced)
//  - 8 rows per wave: indices come in via one scalar load burst (SMEM),
//    8 independent global_load_b128 gathers in flight, then 8 non-temporal
//    global_store_b128 so the 335MB streamed output doesn't evict reused
//    table rows from the 192MB L2 (indices repeat ~2.4x per table).

typedef __attribute__((ext_vector_type(4))) float v4f;

constexpr int      kTables      = 8;
constexpr int      kVocab       = 200000;
constexpr int      kDim         = 128;        // 32 lanes * 4 floats
constexpr unsigned kNNZ         = 81920;
constexpr int      kWavesPerBlk = 8;          // 256 threads = 8 wave32
constexpr int      kRowsPerWave = 8;          // 8 b128 loads in flight/wave
constexpr int      kBlock       = kWavesPerBlk * 32;

__global__ __launch_bounds__(kBlock) void emb_gather_kernel(
    const float* __restrict__ tables,
    const int*   __restrict__ indices,
    float*       __restrict__ out)
{
    const unsigned lane = threadIdx.x & 31u;
    // Lane-uniform wave id -> keep all row/address bases in SGPRs.
    const unsigned wave =
        (unsigned)__builtin_amdgcn_readfirstlane((int)(threadIdx.x >> 5));

    // Exact grid: no bounds check needed. row0 max = 655352, fits u32.
    const unsigned group = blockIdx.x * kWavesPerBlk + wave;
    const unsigned row0  = group * kRowsPerWave;

    // NNZ % kRowsPerWave == 0 -> an 8-row group never straddles two tables.
    const unsigned t = row0 / kNNZ;
    const float* __restrict__ tbl =
        tables + (size_t)t * (size_t)kVocab * (size_t)kDim;

    // 8 wave-uniform indices -> broadcast to SGPRs (scalar index loads,
    // saddr-form row gathers). Unsigned to avoid sign-extend chains.
    unsigned idx[kRowsPerWave];
#pragma unroll
    for (int r = 0; r < kRowsPerWave; ++r)
        idx[r] = (unsigned)__builtin_amdgcn_readfirstlane(indices[row0 + r]);

    // 8 independent coalesced 512B row gathers in flight (temporal policy:
    // let L2 keep hot rows).
    v4f val[kRowsPerWave];
#pragma unroll
    for (int r = 0; r < kRowsPerWave; ++r) {
        const v4f* __restrict__ src =
            (const v4f*)(tbl + (size_t)idx[r] * kDim + lane * 4u);
        val[r] = *src;
    }

    // Streamed output: non-temporal b128 stores (don't pollute L2).
    float* __restrict__ outRow = out + (size_t)row0 * kDim + lane * 4u;
#pragma unroll
    for (int r = 0; r < kRowsPerWave; ++r) {
        __builtin_nontemporal_store(val[r], (v4f*)(outRow + (size_t)r * kDim));
    }
}

extern "C" void kernel_launch(void* const* d_in, const int* in_sizes, int n_in,
                              void* d_out, int out_size, void* d_ws, size_t ws_size,
                              hipStream_t stream) {
    (void)in_sizes; (void)n_in; (void)out_size; (void)d_ws; (void)ws_size;

    const float* tables  = (const float*)d_in[0];
    const int*   indices = (const int*)d_in[1];
    float*       out     = (float*)d_out;

    // 8 * 81920 rows, 8 rows/wave, 8 waves/block -> exact grid (10240 blocks).
    const unsigned totalRows = (unsigned)kTables * kNNZ;          // 655360
    const unsigned groups    = totalRows / kRowsPerWave;          // 81920
    const unsigned blocks    = groups / kWavesPerBlk;             // 10240

    emb_gather_kernel<<<blocks, kBlock, 0, stream>>>(tables, indices, out);
}